// NonNegLinear_6408091205879
// MI455X (gfx1250) — compile-verified
//
#include <hip/hip_runtime.h>

// Problem constants (from reference setup_inputs)
#define B_ 128
#define P_ 1024
#define C_ 1000
#define C4_ 250   // C_/4, exact

typedef __attribute__((ext_vector_type(2))) float v2f;
typedef __attribute__((ext_vector_type(4))) float f4;
typedef __attribute__((ext_vector_type(8))) float v8f;

// ---------------------------------------------------------------------------
// Kernel 1: wT[p][c] = clamp(weight[c][p], 0, 5)   (LDS-tiled 32x32 transpose)
// 4 MB total -> negligible vs the 525 MB main stream; keeps both sides coalesced.
// ---------------------------------------------------------------------------
__global__ __launch_bounds__(256)
void transpose_clip_kernel(const float* __restrict__ w, float* __restrict__ wT) {
    __shared__ float tile[32][33];           // +1 pad: no LDS bank conflicts
    const int pT = blockIdx.x * 32;          // p tile base (P_=1024 -> always in range)
    const int cT = blockIdx.y * 32;          // c tile base (C_=1000 -> tail-checked)
    const int tx = threadIdx.x;              // 0..31
    const int ty = threadIdx.y;              // 0..7
#pragma unroll
    for (int i = 0; i < 32; i += 8) {
        const int c = cT + ty + i;
        const int p = pT + tx;               // coalesced read along p
        float v = 0.0f;
        if (c < C_) v = w[c * P_ + p];
        tile[ty + i][tx] = fminf(fmaxf(v, 0.0f), 5.0f);
    }
    __syncthreads();
#pragma unroll
    for (int i = 0; i < 32; i += 8) {
        const int p = pT + ty + i;
        const int c = cT + tx;               // coalesced write along c
        if (c < C_) wT[p * C_ + c] = tile[tx][ty + i];
    }
}

// ---------------------------------------------------------------------------
// Kernel 2: importance[b][p][c] = input[b][p] * wT[p][c]   (the 525 MB stream)
// One block = one (b, 8-row p-slab). input[b,p] is block-uniform -> scalar load.
// wT rows are read as b128 (L2-resident, reused B_ times); importance is written
// with NON-TEMPORAL b128 stores so the output stream doesn't evict wT from L2.
// Store-bandwidth bound: ~524.8MB / 23.3 TB/s ~= 22.5 us floor.
// ---------------------------------------------------------------------------
#define PP 8
__global__ __launch_bounds__(256)
void importance_kernel(const float* __restrict__ inp,
                       const float* __restrict__ wT,
                       float* __restrict__ out) {
    const int b  = blockIdx.y;
    const int p0 = blockIdx.x * PP;
    const int t  = threadIdx.x;
    if (t >= C4_) return;                    // 250 active lanes per block

    const f4* __restrict__ wT4 = (const f4*)wT;     // 1000 floats = 250 f4 / row
    f4* __restrict__ out4 = (f4*)out;
    const long long base = ((long long)b * P_ + p0) * C4_ + t;

#pragma unroll
    for (int i = 0; i < PP; ++i) {
        const int p = p0 + i;
        const float s = inp[b * P_ + p];     // uniform -> s_load
        f4 r = wT4[p * C4_ + t] * s;
        // threshold (where(v<0, 0, v)) -- free, kernel is store-bound
        r.x = fmaxf(r.x, 0.0f); r.y = fmaxf(r.y, 0.0f);
        r.z = fmaxf(r.z, 0.0f); r.w = fmaxf(r.w, 0.0f);
        __builtin_nontemporal_store(r, &out4[base + (long long)i * C4_]);
    }
}

// ---------------------------------------------------------------------------
// Kernel 3: logits[b][c] = sum_p input[b][p]*wT[p][c] + bias[c]
// 128x1000x1024 fp32 GEMM via V_WMMA_F32_16X16X4_F32. One wave per 16x16 tile,
// 256 WMMAs over K=1024. Column index is clamped for loads so EXEC stays all-1s
// through the WMMA loop (ISA requirement); the C=1000 tail is handled by a
// predicated store after the loop.
// A layout (16x4 f32): lane m=lane&15, vgpr0/1 hold K = k+2*hi, k+2*hi+1.
// B layout (4x16 f32): lane n=lane&15, vgpr0/1 hold the same K rows.
// D layout: vgpr r -> row r + 8*hi, col n.
// ---------------------------------------------------------------------------
__global__ __launch_bounds__(32)
void logits_wmma_kernel(const float* __restrict__ inp,
                        const float* __restrict__ wT,
                        const float* __restrict__ bias,
                        float* __restrict__ logits) {
    const int n0   = blockIdx.x * 16;        // column tile (63 tiles cover 1008)
    const int m0   = blockIdx.y * 16;        // row tile (8 tiles, exact)
    const int lane = threadIdx.x;
    const int n    = lane & 15;
    const int hi   = lane >> 4;
    const int col  = n0 + n;
    const int colc = (col < C_) ? col : (C_ - 1);   // clamp: keep EXEC all-1s

    const float* __restrict__ arow = inp + (m0 + n) * P_;   // A: lane holds row m=n

    v8f acc = {};
#pragma unroll 8
    for (int k = 0; k < P_; k += 4) {
        const int kb = k + 2 * hi;           // even -> a.{x,y} merge into b64 load
        v2f a;  a.x = arow[kb];              a.y = arow[kb + 1];
        v2f bm; bm.x = wT[kb * C_ + colc];   bm.y = wT[(kb + 1) * C_ + colc];
        acc = __builtin_amdgcn_wmma_f32_16x16x4_f32(
            /*neg_a=*/false, a, /*neg_b=*/false, bm,
            /*c_mod=*/(short)0, acc, /*reuse_a=*/false, /*reuse_b=*/false);
    }

    if (col < C_) {
        const float bv = bias[col];
#pragma unroll
        for (int r = 0; r < 8; ++r) {
            const int row = m0 + r + 8 * hi;
            logits[row * C_ + col] = acc[r] + bv;
        }
    }
}

// ---------------------------------------------------------------------------
extern "C" void kernel_launch(void* const* d_in, const int* in_sizes, int n_in,
                              void* d_out, int out_size, void* d_ws, size_t ws_size,
                              hipStream_t stream) {
    const float* inp  = (const float*)d_in[0];   // [B,P]
    const float* w    = (const float*)d_in[1];   // [C,P]
    const float* bias = (const float*)d_in[2];   // [C]

    float* out    = (float*)d_out;                           // importance [B,P,C]
    float* logits = out + (long long)B_ * P_ * C_;           // logits [B,C]
    float* wT     = (float*)d_ws;                            // [P,C] = 4.096 MB

    // 1) clamp + transpose weight into workspace
    transpose_clip_kernel<<<dim3(P_ / 32, (C_ + 31) / 32), dim3(32, 8), 0, stream>>>(w, wT);

    // 2) stream the 525 MB importance tensor (NT b128 stores)
    importance_kernel<<<dim3(P_ / PP, B_), 256, 0, stream>>>(inp, wT, out);

    // 3) tiny WMMA GEMM for logits (+bias)
    logits_wmma_kernel<<<dim3((C_ + 15) / 16, B_ / 16), 32, 0, stream>>>(inp, wT, bias, logits);
}